// ElasticTransform_5574867550531
// MI455X (gfx1250) — compile-verified
//
#include <hip/hip_runtime.h>
#include <math.h>

typedef __attribute__((ext_vector_type(2))) float v2f;
typedef __attribute__((ext_vector_type(8))) float v8f;

#define SIGMA   5.0f
#define ALPHA   50.0f
#define RADIUS  20
#define KSIZE   41            // 2*RADIUS+1 taps
#define KOFF    15            // guard offset: index d in [-15,56] -> kpad[KOFF+d]
#define KPAD    72            // zero-padded tap table
#define IMG_H   512
#define IMG_W   512
#define BATCH   64
#define TILES_X (IMG_W / 16)
#define TILES_Y (IMG_H / 16)
#define WPB     4             // waves per block (128 threads)
#define MTILE   4             // 16x16 tiles per wave (strip processing)
#define NSTEP   14            // 56-wide band / K=4 per V_WMMA_F32_16X16X4_F32
#define HBAND   (16 * MTILE + 2 * RADIUS)   // 104 staged columns (h-pass)
#define HPITCH  105           // gcd(105,64)=1 -> conflict-free LDS rows

// scipy 'reflect' == jnp.pad 'symmetric': (d c b a | a b c d | d c b a)
__device__ __forceinline__ int reflect_idx(int i, int n) {
    int p = i % (2 * n);
    if (p < 0) p += 2 * n;
    return (p < n) ? p : (2 * n - 1 - p);
}

// Normalized Gaussian taps with zero guard bands -> branchless fragment builds.
__device__ __forceinline__ void build_kpad(float* kpad) {
    int t = threadIdx.x;
    if (t < KPAD) {
        float v = 0.0f;
        if (t >= KOFF && t < KOFF + KSIZE) {
            float s = 0.0f;
            for (int i = 0; i < KSIZE; ++i) {
                float d = (float)(i - RADIUS) / SIGMA;
                s += __expf(-0.5f * d * d);
            }
            float d = (float)(t - KOFF - RADIUS) / SIGMA;
            v = __expf(-0.5f * d * d) / s;
        }
        kpad[t] = v;
    }
}

// Horizontal blur of (2*rnd - 1). One 16x64 strip (4 tiles) per wave.
// out[y0+m][x0+16t+j] = sum_c in[y0+m][x0-20+16t+c] * kern[c-j]
__global__ void __launch_bounds__(128) eltx_blur_h(const float* __restrict__ rnd,
                                                   float* __restrict__ out) {
    __shared__ float kpad[KPAD];
    __shared__ float tile[WPB][16][HPITCH];
    build_kpad(kpad);

    const int lane = threadIdx.x & 31;
    const int wave = threadIdx.x >> 5;
    const int gid  = blockIdx.x * WPB + wave;
    const int GX   = TILES_X / MTILE;
    const int img  = gid / (TILES_Y * GX);
    const int rem  = gid - img * (TILES_Y * GX);
    const int ty   = rem / GX;
    const int gx   = rem - ty * GX;
    const int y0   = ty * 16;
    const int x0   = gx * (16 * MTILE);
    const float* src = rnd + (size_t)img * IMG_H * IMG_W;

    // Stage 16 x 104 band: reflect each lane's columns once, fuse 2r-1.
    for (int c = lane; c < HBAND; c += 32) {
        int xc = reflect_idx(x0 - RADIUS + c, IMG_W);
#pragma unroll
        for (int r = 0; r < 16; ++r)
            tile[wave][r][c] = 2.0f * src[(size_t)(y0 + r) * IMG_W + xc] - 1.0f;
    }
    __syncthreads();

    // B fragments (banded kernel matrix) -- identical for all 4 sub-tiles.
    const int j    = lane & 15;
    const int ksel = (lane < 16) ? 0 : 2;     // K split across lane halves
    v2f bfrag[NSTEP];
#pragma unroll
    for (int s = 0; s < NSTEP; ++s) {
        int c0 = 4 * s + ksel;
        bfrag[s].x = kpad[KOFF + c0 - j];     // zero outside the 41-tap window
        bfrag[s].y = kpad[KOFF + c0 + 1 - j];
    }

    const int m      = lane & 15;
    const int rowoff = (lane < 16) ? 0 : 8;
    float* dst = out + (size_t)img * IMG_H * IMG_W;

#pragma unroll
    for (int t = 0; t < MTILE; ++t) {
        v8f acc = {};
#pragma unroll
        for (int s = 0; s < NSTEP; ++s) {
            int c0 = 16 * t + 4 * s + ksel;
            v2f a;
            a.x = tile[wave][m][c0];
            a.y = tile[wave][m][c0 + 1];
            acc = __builtin_amdgcn_wmma_f32_16x16x4_f32(false, a, false, bfrag[s],
                                                        (short)0, acc, false, false);
        }
#pragma unroll
        for (int r = 0; r < 8; ++r)
            dst[(size_t)(y0 + r + rowoff) * IMG_W + x0 + 16 * t + j] = acc[r];
    }
}

// Vertical blur, fused *ALPHA. One 64x16 strip (4 tiles) per wave with a
// rolling register window of B fragments: window slides 16 rows = 4 k-steps,
// so only 4 of 14 fragments are reloaded per tile.
__global__ void __launch_bounds__(128) eltx_blur_v(const float* __restrict__ in,
                                                   float* __restrict__ out) {
    __shared__ float kpad[KPAD];
    build_kpad(kpad);
    __syncthreads();

    const int lane = threadIdx.x & 31;
    const int wave = threadIdx.x >> 5;
    const int gid  = blockIdx.x * WPB + wave;
    const int GY   = TILES_Y / MTILE;
    const int img  = gid / (GY * TILES_X);
    const int rem  = gid - img * (GY * TILES_X);
    const int gy   = rem / TILES_X;
    const int tx   = rem - gy * TILES_X;
    const int y0   = gy * (16 * MTILE);
    const int x0   = tx * 16;
    const float* src = in + (size_t)img * IMG_H * IMG_W;

    const int m    = lane & 15;
    const int ksel = (lane < 16) ? 0 : 2;

    // A fragments: transposed kernel band, loop-invariant across the strip.
    v2f afrag[NSTEP];
#pragma unroll
    for (int s = 0; s < NSTEP; ++s) {
        int c0 = 4 * s + ksel;
        afrag[s].x = kpad[KOFF + c0 - m];
        afrag[s].y = kpad[KOFF + c0 + 1 - m];
    }

    // Prime the rolling B window for tile 0 (rows y0-20 .. y0+35).
    v2f barr[NSTEP];
#pragma unroll
    for (int s = 0; s < NSTEP; ++s) {
        int r0 = y0 - RADIUS + 4 * s + ksel;
        barr[s].x = src[(size_t)reflect_idx(r0,     IMG_H) * IMG_W + x0 + m];
        barr[s].y = src[(size_t)reflect_idx(r0 + 1, IMG_H) * IMG_W + x0 + m];
    }

    float* dst = out + (size_t)img * IMG_H * IMG_W;
    const int rowoff = (lane < 16) ? 0 : 8;

#pragma unroll
    for (int t = 0; t < MTILE; ++t) {
        v8f acc = {};
#pragma unroll
        for (int s = 0; s < NSTEP; ++s)
            acc = __builtin_amdgcn_wmma_f32_16x16x4_f32(false, afrag[s], false, barr[s],
                                                        (short)0, acc, false, false);
        const int yt = y0 + 16 * t;
#pragma unroll
        for (int r = 0; r < 8; ++r)
            dst[(size_t)(yt + r + rowoff) * IMG_W + x0 + m] = ALPHA * acc[r];

        if (t < MTILE - 1) {
#pragma unroll
            for (int s = 0; s < NSTEP - 4; ++s) barr[s] = barr[s + 4];
#pragma unroll
            for (int s = NSTEP - 4; s < NSTEP; ++s) {
                int r0 = (yt + 16) - RADIUS + 4 * s + ksel;
                barr[s].x = src[(size_t)reflect_idx(r0,     IMG_H) * IMG_W + x0 + m];
                barr[s].y = src[(size_t)reflect_idx(r0 + 1, IMG_H) * IMG_W + x0 + m];
            }
        }
    }
}

// Bilinear warp with reflect boundary. One thread per output pixel.
__global__ void __launch_bounds__(256) eltx_warp(const float* __restrict__ x,
                                                 const float* __restrict__ dxF,
                                                 const float* __restrict__ dyF,
                                                 float* __restrict__ out) {
    const int idx = blockIdx.x * 256 + threadIdx.x;     // total is exact multiple
    const int img = idx / (IMG_H * IMG_W);
    const int rem = idx - img * (IMG_H * IMG_W);
    const int yi  = rem / IMG_W;
    const int xi  = rem - yi * IMG_W;

    float xx  = (float)xi + dxF[idx];
    float yy  = (float)yi + dyF[idx];
    float x0f = floorf(xx), y0f = floorf(yy);
    float wx  = xx - x0f,   wy  = yy - y0f;
    int x0i = (int)x0f, y0i = (int)y0f;
    int x0r = reflect_idx(x0i,     IMG_W);
    int x1r = reflect_idx(x0i + 1, IMG_W);
    int y0r = reflect_idx(y0i,     IMG_H);
    int y1r = reflect_idx(y0i + 1, IMG_H);

    const float* p = x + (size_t)img * IMG_H * IMG_W;
    float v00 = p[(size_t)y0r * IMG_W + x0r];
    float v01 = p[(size_t)y0r * IMG_W + x1r];
    float v10 = p[(size_t)y1r * IMG_W + x0r];
    float v11 = p[(size_t)y1r * IMG_W + x1r];

    out[idx] = v00 * (1.0f - wy) * (1.0f - wx) + v01 * (1.0f - wy) * wx
             + v10 * wy * (1.0f - wx)          + v11 * wy * wx;
}

extern "C" void kernel_launch(void* const* d_in, const int* in_sizes, int n_in,
                              void* d_out, int out_size, void* d_ws, size_t ws_size,
                              hipStream_t stream) {
    (void)in_sizes; (void)n_in; (void)out_size; (void)ws_size;
    const float* x   = (const float*)d_in[0];
    const float* rdx = (const float*)d_in[1];
    const float* rdy = (const float*)d_in[2];
    float* out = (float*)d_out;

    const size_t field = (size_t)BATCH * IMG_H * IMG_W;
    float* tmp = (float*)d_ws;          // one field of scratch, reused per channel
    float* dxF = tmp + field;
    float* dyF = dxF + field;

    const unsigned grpH = (unsigned)(BATCH * TILES_Y * (TILES_X / MTILE) / WPB);
    const unsigned grpV = (unsigned)(BATCH * (TILES_Y / MTILE) * TILES_X / WPB);
    dim3 blk(128);

    eltx_blur_h<<<dim3(grpH), blk, 0, stream>>>(rdx, tmp);
    eltx_blur_v<<<dim3(grpV), blk, 0, stream>>>(tmp, dxF);
    eltx_blur_h<<<dim3(grpH), blk, 0, stream>>>(rdy, tmp);
    eltx_blur_v<<<dim3(grpV), blk, 0, stream>>>(tmp, dyF);
    eltx_warp<<<dim3((unsigned)(field / 256)), dim3(256), 0, stream>>>(x, dxF, dyF, out);
}